// CenterLoss_3882650436548
// MI455X (gfx1250) — compile-verified
//
#include <hip/hip_runtime.h>

typedef __attribute__((ext_vector_type(2))) float v2f;
typedef __attribute__((ext_vector_type(8))) float v8f;

#define CLAMP_MIN 1e-12f
#define CLAMP_MAX 1e12f

// Each wave (32 lanes) handles 16 samples. A-matrix (16x4 f32) layout:
//   lanes 0-15  : row M = lane,    K = {k0, k0+1}  (VGPR0,VGPR1)
//   lanes 16-31 : row M = lane-16, K = {k0+2, k0+3}
// We put diff^2 in A and multiply by an all-ones B (4x16), so
//   D[m,n] = sum_k diff^2[m,k]  for every n  (row sums, replicated per column).
// C/D layout: VGPR v, lanes 0-15 -> M=v; lanes 16-31 -> M=v+8. Since all
// columns are identical, lane 0 holds rows 0..7 and lane 16 holds rows 8..15.
__global__ __launch_bounds__(256) void center_dist_wmma_kernel(
    const float* __restrict__ x, const int* __restrict__ labels,
    const float* __restrict__ centers, float* __restrict__ ws, int Dn) {
  const int tid  = threadIdx.x;
  const int lane = tid & 31;
  const int wave = tid >> 5;
  const int r    = lane & 15;   // sample row within the wave's 16-sample tile
  const int half = lane >> 4;   // 0: K{0,1}, 1: K{2,3}
  const int base = blockIdx.x * 128 + wave * 16;
  const int row  = base + r;

  const int lab = labels[row];
  const float* __restrict__ xr = x + (long)row * Dn;
  const float* __restrict__ cr = centers + (long)lab * Dn;

  v2f ones; ones[0] = 1.0f; ones[1] = 1.0f;
  v8f acc = {};

  const int k0 = half * 2;
#pragma unroll 4
  for (int j = 0; j < Dn; j += 4) {
    v2f xv = *(const v2f*)(xr + j + k0);
    v2f cv = *(const v2f*)(cr + j + k0);
    v2f dv = xv - cv;
    dv = dv * dv;  // A[m,k] = diff^2 so B=ones yields row sums of squares
    // 8 args: (neg_a, A, neg_b, B, c_mod, C, reuse_a, reuse_b)
    acc = __builtin_amdgcn_wmma_f32_16x16x4_f32(
        false, dv, false, ones, (short)0, acc, false, false);
  }

  if (lane == 0 || lane == 16) {
    const int roff = base + half * 8;
#pragma unroll
    for (int v = 0; v < 8; ++v) {
      float d = acc[v];
      d = fminf(fmaxf(d, CLAMP_MIN), CLAMP_MAX);
      ws[roff + v] = d;
    }
  }
}

// Deterministic single-block mean reduction (fixed per-thread order + fixed
// tree): bitwise-identical result on every replay, no float atomics.
__global__ __launch_bounds__(256) void reduce_mean_kernel(
    const float* __restrict__ ws, float* __restrict__ out, int Bn) {
  __shared__ float sm[256];
  const int tid = threadIdx.x;
  float s = 0.0f;
  for (int i = tid; i < Bn; i += 256) s += ws[i];
  sm[tid] = s;
  __syncthreads();
#pragma unroll
  for (int off = 128; off > 0; off >>= 1) {
    if (tid < off) sm[tid] += sm[tid + off];
    __syncthreads();
  }
  if (tid == 0) out[0] = sm[0] * (1.0f / (float)Bn);
}

extern "C" void kernel_launch(void* const* d_in, const int* in_sizes, int n_in,
                              void* d_out, int out_size, void* d_ws, size_t ws_size,
                              hipStream_t stream) {
  const float* x       = (const float*)d_in[0];
  const int*   labels  = (const int*)d_in[1];
  const float* centers = (const float*)d_in[2];

  const int Bn = in_sizes[1];             // 8192 samples
  const int Dn = in_sizes[0] / Bn;        // 512 features

  float* ws = (float*)d_ws;               // Bn floats of per-sample distance

  const int samples_per_block = 128;      // 8 waves x 16 samples
  const int grid = Bn / samples_per_block;

  center_dist_wmma_kernel<<<grid, 256, 0, stream>>>(x, labels, centers, ws, Dn);
  reduce_mean_kernel<<<1, 256, 0, stream>>>(ws, (float*)d_out, Bn);
}